// GINE_Net_64888365908462
// MI455X (gfx1250) — compile-verified
//
#include <hip/hip_runtime.h>
#include <hip/hip_bf16.h>

#define HID 128
#define NLAYER 4
#define NN 50000
#define NE 600000
#define NG 256
#define AF 9
#define AV 64
#define BF 3
#define BV 8
#define BN_EPS 1e-5f

typedef __attribute__((ext_vector_type(2))) float v2f;
typedef __attribute__((ext_vector_type(8))) float v8f;

// ---------------- embeddings ----------------
__global__ void embed_nodes_k(const int* __restrict__ x,
                              const float* __restrict__ emb,
                              float* __restrict__ h) {
  int gid = blockIdx.x * blockDim.x + threadIdx.x;
  if (gid >= NN * HID) return;
  int n = gid >> 7, c = gid & 127;
  float acc = 0.f;
#pragma unroll
  for (int f = 0; f < AF; ++f) {
    int v = x[n * AF + f];
    acc += emb[(f * AV + v) * HID + c];
  }
  h[gid] = acc;
}

__global__ void embed_edges_k(const int* __restrict__ ea,
                              const float* __restrict__ emb,
                              float* __restrict__ e) {
  int gid = blockIdx.x * blockDim.x + threadIdx.x;
  if (gid >= NE * HID) return;
  int j = gid >> 7, c = gid & 127;
  float acc = 0.f;
#pragma unroll
  for (int f = 0; f < BF; ++f) {
    int v = ea[j * BF + f];
    acc += emb[(f * BV + v) * HID + c];
  }
  e[gid] = acc;
}

// ---------------- utility ----------------
__global__ void copy4_k(const float4* __restrict__ s, float4* __restrict__ d, int n4) {
  int i = blockIdx.x * blockDim.x + threadIdx.x;
  if (i < n4) d[i] = s[i];
}

__global__ void zero_k(float* __restrict__ p, int n) {
  int i = blockIdx.x * blockDim.x + threadIdx.x;
  if (i < n) p[i] = 0.f;
}

// ---------------- pack all weight matrices into WMMA B-fragment order ----------
// frag(n,k) = 64 contiguous floats; element for lane L (=half*16+l15), pair e:
//   wf[mat][ ((n*32+k)*32 + L)*2 + e ] = W[mat][4k + 2*half + e][16n + l15]
__global__ void pack_w_k(const float* __restrict__ lin1_w,
                         const float* __restrict__ lin2_w,
                         float* __restrict__ wf) {
  int gid = blockIdx.x * blockDim.x + threadIdx.x;
  const int per = HID * HID;
  if (gid >= 2 * NLAYER * per) return;
  int mat = gid / per;
  int rem = gid - mat * per;
  int r = rem >> 7;        // input dim (K)
  int c = rem & 127;       // output dim (N)
  const float* src = (mat < NLAYER) ? (lin1_w + (size_t)mat * per)
                                    : (lin2_w + (size_t)(mat - NLAYER) * per);
  float v = src[rem];
  int k = r >> 2, half = (r >> 1) & 1, e = r & 1;
  int n = c >> 4, l15 = c & 15;
  int dst = ((n * 32 + k) * 32 + half * 16 + l15) * 2 + e;
  wf[(size_t)mat * per + dst] = v;
}

// ---------------- message scatter: zin[dst] += relu(h[src] + e) ----------------
__global__ void scatter_msg_k(const int* __restrict__ ei,
                              const float* __restrict__ h,
                              const float* __restrict__ e,
                              float* __restrict__ zin) {
  int gid = blockIdx.x * blockDim.x + threadIdx.x;
  if (gid >= NE * 32) return;
  int j = gid >> 5;
  int c4 = (gid & 31) << 2;
  int s = ei[j];
  int d = ei[NE + j];
  if (j + 512 < NE) __builtin_prefetch(&e[(size_t)(j + 512) * HID + c4], 0, 1);
  const float4 hv = *(const float4*)&h[(size_t)s * HID + c4];
  const float4 ev = *(const float4*)&e[(size_t)j * HID + c4];
  float4 m;
  m.x = fmaxf(hv.x + ev.x, 0.f);
  m.y = fmaxf(hv.y + ev.y, 0.f);
  m.z = fmaxf(hv.z + ev.z, 0.f);
  m.w = fmaxf(hv.w + ev.w, 0.f);
  float* zp = &zin[(size_t)d * HID + c4];
  atomicAdd(zp + 0, m.x);
  atomicAdd(zp + 1, m.y);
  atomicAdd(zp + 2, m.z);
  atomicAdd(zp + 3, m.w);
}

// ---------------- fused GINE MLP layer: WMMA f32 16x16x4 ----------------
// h_out = h_in + relu(BN(relu(zin @ W1 + b1) @ W2 + b2))
#define LDA 132   // padded LDS row stride for A/T tiles
#define BROWS 128
#define WFSZ (HID * HID)   // 16384 floats, packed fragment array

__global__ __launch_bounds__(256) void gine_mlp_k(
    const float* __restrict__ zin, const float* __restrict__ hin,
    float* __restrict__ hout,
    const float* __restrict__ w1f, const float* __restrict__ b1,
    const float* __restrict__ w2f, const float* __restrict__ b2,
    const float* __restrict__ gamma, const float* __restrict__ beta,
    const float* __restrict__ mean, const float* __restrict__ var) {
  extern __shared__ float smem[];
  float* lds_a  = smem;                         // 128 x LDA : input tile
  float* lds_t  = smem + BROWS * LDA;           // 128 x LDA : relu(GEMM1)
  float* lds_wf = smem + 2 * BROWS * LDA;       // 16384     : packed W frags

  const int tid  = threadIdx.x;
  const int wave = tid >> 5;
  const int lane = tid & 31;
  const int half = lane >> 4;
  const int l15  = lane & 15;
  const int rowbase = blockIdx.x * BROWS;
  const int mloc = wave * 16;

  // cooperative stage: A tile (row-clamped) + packed W1 (plain coalesced copy)
#pragma unroll
  for (int it = 0; it < 16; ++it) {
    int f = it * 256 + tid;       // float4 index
    int r = f >> 5;
    int c = (f & 31) << 2;
    int gr = rowbase + r;
    if (gr >= NN) gr = NN - 1;
    *(float4*)&lds_a[r * LDA + c] = *(const float4*)&zin[(size_t)gr * HID + c];
    *(float4*)&lds_wf[f << 2]     = *(const float4*)&w1f[f << 2];
  }
  __syncthreads();

  // A fragments: a[k] covers K = 4k .. 4k+3 (contiguous 8B-aligned pair ->
  // single ds_load_b64 with a 16-bit immediate offset)
  const float* abase = &lds_a[(mloc + l15) * LDA + 2 * half];
  v2f afrag[32];
#pragma unroll
  for (int k = 0; k < 32; ++k)
    afrag[k] = *(const v2f*)(abase + 4 * k);

  const float* wlane = &lds_wf[lane * 2];   // one base; imm offsets up to 65280B

  // GEMM1 -> +b1 -> relu -> lds_t
  for (int n = 0; n < 8; ++n) {
    const int cn = 16 * n + l15;
    v2f bfrag[32];
#pragma unroll
    for (int k = 0; k < 32; ++k)
      bfrag[k] = *(const v2f*)(wlane + (n * 32 + k) * 64);
    v8f acc = {};
#pragma unroll
    for (int k = 0; k < 32; ++k)
      acc = __builtin_amdgcn_wmma_f32_16x16x4_f32(false, afrag[k], false, bfrag[k],
                                                  (short)0, acc, false, false);
    float bc = b1[cn];
#pragma unroll
    for (int r = 0; r < 8; ++r) {
      float v = fmaxf(acc[r] + bc, 0.f);
      lds_t[(mloc + r + 8 * half) * LDA + cn] = v;
    }
  }
  __syncthreads();

  // stage packed W2 over W1
#pragma unroll
  for (int it = 0; it < 16; ++it) {
    int f = it * 256 + tid;
    *(float4*)&lds_wf[f << 2] = *(const float4*)&w2f[f << 2];
  }
  __syncthreads();

  // T fragments (same contiguous-pair b64 form)
  const float* tbase = &lds_t[(mloc + l15) * LDA + 2 * half];
#pragma unroll
  for (int k = 0; k < 32; ++k)
    afrag[k] = *(const v2f*)(tbase + 4 * k);

  // GEMM2 -> +b2 -> BN -> relu -> residual
  for (int n = 0; n < 8; ++n) {
    const int cn = 16 * n + l15;
    v2f bfrag[32];
#pragma unroll
    for (int k = 0; k < 32; ++k)
      bfrag[k] = *(const v2f*)(wlane + (n * 32 + k) * 64);
    v8f acc = {};
#pragma unroll
    for (int k = 0; k < 32; ++k)
      acc = __builtin_amdgcn_wmma_f32_16x16x4_f32(false, afrag[k], false, bfrag[k],
                                                  (short)0, acc, false, false);
    float kk = gamma[cn] * rsqrtf(var[cn] + BN_EPS);
    float sh = beta[cn] - mean[cn] * kk;
    float bc = b2[cn];
#pragma unroll
    for (int r = 0; r < 8; ++r) {
      int gr = rowbase + mloc + r + 8 * half;
      if (gr < NN) {
        float z = (acc[r] + bc) * kk + sh;
        z = fmaxf(z, 0.f);
        hout[(size_t)gr * HID + cn] = hin[(size_t)gr * HID + cn] + z;
      }
    }
  }
}

// ---------------- pooling + output MLP ----------------
__global__ void pool_scatter_k(const float* __restrict__ h,
                               const int* __restrict__ batch,
                               float* __restrict__ sums,
                               float* __restrict__ cnt) {
  int gid = blockIdx.x * blockDim.x + threadIdx.x;
  if (gid >= NN * HID) return;
  int n = gid >> 7, c = gid & 127;
  int b = batch[n];
  atomicAdd(&sums[b * HID + c], h[gid]);
  if (c == 0) atomicAdd(&cnt[b], 1.f);
}

__global__ void pool_mlp_k(const float* __restrict__ sums,
                           const float* __restrict__ cnt,
                           const float* __restrict__ mw,
                           const float* __restrict__ mb,
                           float* __restrict__ out) {
  int g = blockIdx.x * blockDim.x + threadIdx.x;
  if (g >= NG) return;
  float cg = fmaxf(cnt[g], 1.f);
  float acc = mb[0];
#pragma unroll
  for (int c = 0; c < HID; ++c) acc += (sums[g * HID + c] / cg) * mw[c];
  out[g] = acc;
}

// ---------------- host launcher ----------------
extern "C" void kernel_launch(void* const* d_in, const int* in_sizes, int n_in,
                              void* d_out, int out_size, void* d_ws, size_t ws_size,
                              hipStream_t stream) {
  (void)in_sizes; (void)n_in; (void)out_size; (void)ws_size;
  const int*   x        = (const int*)  d_in[0];
  const int*   ei       = (const int*)  d_in[1];
  const int*   ea       = (const int*)  d_in[2];
  const int*   batch    = (const int*)  d_in[3];
  const float* atom_emb = (const float*)d_in[4];
  const float* bond_emb = (const float*)d_in[5];
  const float* lin1_w   = (const float*)d_in[6];
  const float* lin1_b   = (const float*)d_in[7];
  const float* lin2_w   = (const float*)d_in[8];
  const float* lin2_b   = (const float*)d_in[9];
  const float* bn_g     = (const float*)d_in[10];
  const float* bn_b     = (const float*)d_in[11];
  const float* bn_m     = (const float*)d_in[12];
  const float* bn_v     = (const float*)d_in[13];
  const float* mlp_w    = (const float*)d_in[14];
  const float* mlp_b    = (const float*)d_in[15];
  float* out = (float*)d_out;

  size_t off = 0;
  auto carve = [&](size_t bytes) {
    void* p = (char*)d_ws + off;
    off += (bytes + 255) & ~(size_t)255;
    return p;
  };
  float* h_a  = (float*)carve((size_t)NN * HID * 4);
  float* h_b  = (float*)carve((size_t)NN * HID * 4);
  float* zin  = (float*)carve((size_t)NN * HID * 4);
  float* e    = (float*)carve((size_t)NE * HID * 4);
  float* wf   = (float*)carve((size_t)2 * NLAYER * WFSZ * 4);
  float* sums = (float*)carve((size_t)NG * HID * 4);
  float* cnt  = (float*)carve((size_t)NG * 4);

  const int smem_bytes = (2 * BROWS * LDA + WFSZ) * 4;
  hipFuncSetAttribute((const void*)gine_mlp_k,
                      hipFuncAttributeMaxDynamicSharedMemorySize, smem_bytes);

  pack_w_k<<<(2 * NLAYER * HID * HID + 255) / 256, 256, 0, stream>>>(lin1_w, lin2_w, wf);
  embed_nodes_k<<<(NN * HID + 255) / 256, 256, 0, stream>>>(x, atom_emb, h_a);
  embed_edges_k<<<(NE * HID + 255) / 256, 256, 0, stream>>>(ea, bond_emb, e);

  float* hc = h_a;
  float* hn = h_b;
  for (int l = 0; l < NLAYER; ++l) {
    copy4_k<<<(NN * HID / 4 + 255) / 256, 256, 0, stream>>>(
        (const float4*)hc, (float4*)zin, NN * HID / 4);
    scatter_msg_k<<<(NE * 32 + 255) / 256, 256, 0, stream>>>(ei, hc, e, zin);
    gine_mlp_k<<<(NN + BROWS - 1) / BROWS, 256, smem_bytes, stream>>>(
        zin, hc, hn,
        wf + (size_t)l * WFSZ, lin1_b + (size_t)l * HID,
        wf + (size_t)(NLAYER + l) * WFSZ, lin2_b + (size_t)l * HID,
        bn_g + (size_t)l * HID, bn_b + (size_t)l * HID,
        bn_m + (size_t)l * HID, bn_v + (size_t)l * HID);
    float* t = hc; hc = hn; hn = t;
  }

  zero_k<<<(NG * HID + 255) / 256, 256, 0, stream>>>(sums, NG * HID);
  zero_k<<<1, 256, 0, stream>>>(cnt, NG);
  pool_scatter_k<<<(NN * HID + 255) / 256, 256, 0, stream>>>(hc, batch, sums, cnt);
  pool_mlp_k<<<1, 256, 0, stream>>>(sums, cnt, mlp_w, mlp_b, out);
}